// RecommendationGNN_66056597012846
// MI455X (gfx1250) — compile-verified
//
#include <hip/hip_runtime.h>

// Problem constants (match reference)
#define NU_ 200000
#define NI_ 100000
#define D_  128
#define H_  256
#define O_  128

typedef __attribute__((ext_vector_type(2))) float v2f;
typedef __attribute__((ext_vector_type(8))) float v8f;

// ---------------------------------------------------------------------------
// Degree count: one thread per edge, atomic +1 to both endpoints' degree.
// ---------------------------------------------------------------------------
__global__ __launch_bounds__(256) void deg_kernel(const int* __restrict__ src,
                                                  const int* __restrict__ dst,
                                                  float* __restrict__ deg_u,
                                                  float* __restrict__ deg_i,
                                                  int e) {
    int i = blockIdx.x * 256 + threadIdx.x;
    if (i < e) {
        atomicAdd(&deg_u[src[i]], 1.0f);
        atomicAdd(&deg_i[dst[i]], 1.0f);
    }
}

// In-place inv = 1 / max(deg, 1)
__global__ __launch_bounds__(256) void inv_kernel(float* __restrict__ deg, int n) {
    int i = blockIdx.x * 256 + threadIdx.x;
    if (i < n) deg[i] = 1.0f / fmaxf(deg[i], 1.0f);
}

// Weight transpose: WT[n*K + k] = W[k*N + n]  (tiny matrices, done once/call)
__global__ __launch_bounds__(256) void transpose_kernel(const float* __restrict__ W,
                                                        float* __restrict__ WT,
                                                        int K, int N) {
    int idx = blockIdx.x * 256 + threadIdx.x;
    if (idx < K * N) {
        int k = idx / N;
        int n = idx - k * N;
        WT[(size_t)n * K + k] = W[idx];
    }
}

// ---------------------------------------------------------------------------
// Scatter-add aggregation: one wave (32 lanes) per edge. Each lane handles
// DF/128 float4s, 32 apart: vector gather from xsrc[gidx[e]], atomic f32 add
// into agg[sidx[e]].  DF = 128 or 256.
// ---------------------------------------------------------------------------
template <int DF>
__global__ __launch_bounds__(256) void scatter_kernel(const int* __restrict__ gidx,
                                                      const int* __restrict__ sidx,
                                                      const float* __restrict__ xsrc,
                                                      float* __restrict__ agg,
                                                      int e) {
    int wave = (blockIdx.x * 256 + threadIdx.x) >> 5;
    int lane = threadIdx.x & 31;
    if (wave >= e) return;
    int g = gidx[wave];
    int s = sidx[wave];
    const float4* xs = (const float4*)(xsrc + (size_t)g * DF);
    float* ag = agg + (size_t)s * DF;
#pragma unroll
    for (int c = 0; c < DF / 128; ++c) {
        float4 v = xs[lane + 32 * c];
        int b = (lane + 32 * c) * 4;
        atomicAdd(&ag[b + 0], v.x);
        atomicAdd(&ag[b + 1], v.y);
        atomicAdd(&ag[b + 2], v.z);
        atomicAdd(&ag[b + 3], v.w);
    }
}

// ---------------------------------------------------------------------------
// Fused SAGE GEMM with fp32 WMMA, full-N row stripe per wave:
//   Out[M,N] = relu( (Agg[M,K]*InvDeg[M]) @ Wl[K,N] + Bias[N] + X[M,K] @ Wr[K,N] )
// Each wave owns 16 rows x N cols (NT = N/16 tiles, acc in registers).
// Weights pre-transposed to [N,K] so every A and B fragment is one b64 load.
//
// ISA 16x4 fp32 A layout: lanes 0-15 -> M=lane, K{k0,k0+1}; lanes 16-31 ->
// M=lane-16, K{k0+2,k0+3}. B 4x16 mirrors it (lane half = K-pair, VGPR = K
// within pair, lane&15 = N). C/D: VGPR v -> M = v + (hi?8:0), N = lane&15.
// ---------------------------------------------------------------------------
template <int K, int NT>
__global__ __launch_bounds__(256) void sage_gemm_kernel(
    const float* __restrict__ Agg, const float* __restrict__ InvDeg,
    const float* __restrict__ X, const float* __restrict__ WlT,
    const float* __restrict__ WrT, const float* __restrict__ Bias,
    float* __restrict__ Out, int M) {
    constexpr int N = NT * 16;
    int wave = (blockIdx.x * 256 + threadIdx.x) >> 5;
    if (wave >= (M >> 4)) return;     // uniform per-wave: EXEC stays all-1s

    int lane = threadIdx.x & 31;
    int lm = lane & 15;
    int hi = (lane >> 4) & 1;         // lane half selects K-pair offset (+2)

    int arow = wave * 16 + lm;        // A row for this lane (both halves)
    const float* aggRow = Agg + (size_t)arow * K;
    const float* xRow   = X   + (size_t)arow * K;
    float inv = InvDeg[arow];

    v8f acc[NT];
    v8f zero = {};
#pragma unroll
    for (int t = 0; t < NT; ++t) acc[t] = zero;

    for (int k0 = 0; k0 < K; k0 += 4) {
        int ka = k0 + (hi << 1);      // k0 or k0+2 (8B aligned)

        float2 ta = *(const float2*)(aggRow + ka);
        v2f a1 = { ta.x * inv, ta.y * inv };
        v2f a2 = *(const v2f*)(xRow + ka);

#pragma unroll
        for (int t = 0; t < NT; ++t) {
            int col = t * 16 + lm;
            v2f b1 = *(const v2f*)(WlT + (size_t)col * K + ka);
            v2f b2 = *(const v2f*)(WrT + (size_t)col * K + ka);
            acc[t] = __builtin_amdgcn_wmma_f32_16x16x4_f32(false, a1, false, b1,
                                                           (short)0, acc[t], false, false);
            acc[t] = __builtin_amdgcn_wmma_f32_16x16x4_f32(false, a2, false, b2,
                                                           (short)0, acc[t], false, false);
        }
    }

    int row0 = wave * 16 + (hi ? 8 : 0);
#pragma unroll
    for (int t = 0; t < NT; ++t) {
        int col = t * 16 + lm;
        float bb = Bias[col];
#pragma unroll
        for (int v = 0; v < 8; ++v) {
            float r = fmaxf(acc[t][v] + bb, 0.0f);
            Out[(size_t)(row0 + v) * N + col] = r;
        }
    }
}

// ---------------------------------------------------------------------------
template <int K, int NT>
static inline void launch_gemm(const float* Agg, const float* Inv, const float* X,
                               const float* WlT, const float* WrT, const float* B,
                               float* Out, int M, hipStream_t s) {
    int waves = M / 16;
    int blocks = (waves + 7) / 8;     // 8 row-stripe waves per 256-thread block
    sage_gemm_kernel<K, NT><<<blocks, 256, 0, s>>>(Agg, Inv, X, WlT, WrT, B, Out, M);
}

extern "C" void kernel_launch(void* const* d_in, const int* in_sizes, int n_in,
                              void* d_out, int out_size, void* d_ws, size_t ws_size,
                              hipStream_t stream) {
    (void)n_in; (void)out_size; (void)ws_size;

    const float* x_user = (const float*)d_in[0];
    const float* x_item = (const float*)d_in[1];
    const int*   esrc   = (const int*)d_in[2];
    const int*   edst   = (const int*)d_in[3];
    const float* Wl1_ui = (const float*)d_in[4];
    const float* Wr1_ui = (const float*)d_in[5];
    const float* bl1_ui = (const float*)d_in[6];
    const float* Wl1_iu = (const float*)d_in[7];
    const float* Wr1_iu = (const float*)d_in[8];
    const float* bl1_iu = (const float*)d_in[9];
    const float* Wl2_ui = (const float*)d_in[10];
    const float* Wr2_ui = (const float*)d_in[11];
    const float* bl2_ui = (const float*)d_in[12];
    const float* Wl2_iu = (const float*)d_in[13];
    const float* Wr2_iu = (const float*)d_in[14];
    const float* bl2_iu = (const float*)d_in[15];

    int E = in_sizes[2];

    // Workspace layout (floats, 256B aligned slabs)
    char* ws = (char*)d_ws;
    size_t off = 0;
    auto alloc = [&](size_t nf) -> float* {
        float* p = (float*)(ws + off);
        off += nf * sizeof(float);
        off = (off + 255) & ~(size_t)255;
        return p;
    };
    float* deg_u  = alloc(NU_);                 // -> inv degree (user)
    float* deg_i  = alloc(NI_);                 // -> inv degree (item)
    float* h_user = alloc((size_t)NU_ * H_);
    float* h_item = alloc((size_t)NI_ * H_);
    float* agg_u  = alloc((size_t)NU_ * H_);    // reused layer1 (D) / layer2 (H)
    float* agg_i  = alloc((size_t)NI_ * H_);
    // transposed weights [N,K]
    float* T_l1_ui = alloc((size_t)D_ * H_);
    float* T_r1_ui = alloc((size_t)D_ * H_);
    float* T_l1_iu = alloc((size_t)D_ * H_);
    float* T_r1_iu = alloc((size_t)D_ * H_);
    float* T_l2_ui = alloc((size_t)H_ * O_);
    float* T_r2_ui = alloc((size_t)H_ * O_);
    float* T_l2_iu = alloc((size_t)H_ * O_);
    float* T_r2_iu = alloc((size_t)H_ * O_);

    float* o_user = (float*)d_out;
    float* o_item = (float*)d_out + (size_t)NU_ * O_;

    // ---- weight transposes (tiny) ----
    {
        int n1 = D_ * H_, b1 = (n1 + 255) / 256;
        transpose_kernel<<<b1, 256, 0, stream>>>(Wl1_ui, T_l1_ui, D_, H_);
        transpose_kernel<<<b1, 256, 0, stream>>>(Wr1_ui, T_r1_ui, D_, H_);
        transpose_kernel<<<b1, 256, 0, stream>>>(Wl1_iu, T_l1_iu, D_, H_);
        transpose_kernel<<<b1, 256, 0, stream>>>(Wr1_iu, T_r1_iu, D_, H_);
        int n2 = H_ * O_, b2 = (n2 + 255) / 256;
        transpose_kernel<<<b2, 256, 0, stream>>>(Wl2_ui, T_l2_ui, H_, O_);
        transpose_kernel<<<b2, 256, 0, stream>>>(Wr2_ui, T_r2_ui, H_, O_);
        transpose_kernel<<<b2, 256, 0, stream>>>(Wl2_iu, T_l2_iu, H_, O_);
        transpose_kernel<<<b2, 256, 0, stream>>>(Wr2_iu, T_r2_iu, H_, O_);
    }

    // ---- degrees + reciprocals ----
    hipMemsetAsync(deg_u, 0, (size_t)NU_ * sizeof(float), stream);
    hipMemsetAsync(deg_i, 0, (size_t)NI_ * sizeof(float), stream);
    deg_kernel<<<(E + 255) / 256, 256, 0, stream>>>(esrc, edst, deg_u, deg_i, E);
    inv_kernel<<<(NU_ + 255) / 256, 256, 0, stream>>>(deg_u, NU_);
    inv_kernel<<<(NI_ + 255) / 256, 256, 0, stream>>>(deg_i, NI_);

    int sblocks = (int)(((size_t)E * 32 + 255) / 256);

    // ---- layer 1: aggregate D-dim features ----
    hipMemsetAsync(agg_u, 0, (size_t)NU_ * D_ * sizeof(float), stream);
    hipMemsetAsync(agg_i, 0, (size_t)NI_ * D_ * sizeof(float), stream);
    scatter_kernel<128><<<sblocks, 256, 0, stream>>>(esrc, edst, x_user, agg_i, E);
    scatter_kernel<128><<<sblocks, 256, 0, stream>>>(edst, esrc, x_item, agg_u, E);

    // h_item = relu(mean_i @ Wl1_ui + b + x_item @ Wr1_ui);  h_user analog
    launch_gemm<D_, H_ / 16>(agg_i, deg_i, x_item, T_l1_ui, T_r1_ui, bl1_ui, h_item, NI_, stream);
    launch_gemm<D_, H_ / 16>(agg_u, deg_u, x_user, T_l1_iu, T_r1_iu, bl1_iu, h_user, NU_, stream);

    // ---- layer 2: aggregate H-dim hidden features ----
    hipMemsetAsync(agg_u, 0, (size_t)NU_ * H_ * sizeof(float), stream);
    hipMemsetAsync(agg_i, 0, (size_t)NI_ * H_ * sizeof(float), stream);
    scatter_kernel<256><<<sblocks, 256, 0, stream>>>(esrc, edst, h_user, agg_i, E);
    scatter_kernel<256><<<sblocks, 256, 0, stream>>>(edst, esrc, h_item, agg_u, E);

    launch_gemm<H_, O_ / 16>(agg_i, deg_i, h_item, T_l2_ui, T_r2_ui, bl2_ui, o_item, NI_, stream);
    launch_gemm<H_, O_ / 16>(agg_u, deg_u, h_user, T_l2_iu, T_r2_iu, bl2_iu, o_user, NU_, stream);
}